// HNGCL_51479478010658
// MI455X (gfx1250) — compile-verified
//
#include <hip/hip_runtime.h>
#include <hip/hip_bf16.h>

typedef float v2f __attribute__((ext_vector_type(2)));
typedef float v8f __attribute__((ext_vector_type(8)));

// ---- guaranteed hardware fp32 atomic add (device scope, no return) ----
__device__ __forceinline__ void atomic_add_f32_dev(float* p, float v) {
  asm volatile("global_atomic_add_f32 %0, %1, off scope:SCOPE_DEV"
               :: "v"(p), "v"(v)
               : "memory");
}

// ---------------- degree / normalization ----------------

__global__ void gcn_deg_init(float* __restrict__ deg, int n) {
  int i = blockIdx.x * blockDim.x + threadIdx.x;
  if (i < n) deg[i] = 1.0f;
}

__global__ void gcn_deg_count(const int* __restrict__ dst, float* __restrict__ deg, int E) {
  int e = blockIdx.x * blockDim.x + threadIdx.x;
  if (e < E) atomic_add_f32_dev(&deg[dst[e]], 1.0f);
}

__global__ void gcn_deg_rsqrt(float* __restrict__ deg, int n) {
  int i = blockIdx.x * blockDim.x + threadIdx.x;
  if (i < n) deg[i] = rsqrtf(deg[i]);
}

// ---------------- fp32 WMMA GEMM: H = X @ W ----------------
// Compile-time K (inner dim) and F (out cols). One wave per 16x16 output tile,
// blockDim = (F/16)*32, blockIdx.x = 16-row strip. The 16xK A tile is staged
// in LDS once per block (row padded +4 dwords -> conflict-free ds_load_b64).
template <int K, int F>
__global__ void gcn_gemm_wmma(const float* __restrict__ X, const float* __restrict__ W,
                              float* __restrict__ H) {
  constexpr int PAD = K + 4;           // 132 dwords: banks (4*r + c) % 64 distinct
  __shared__ float As[16 * PAD];

  const int wave = threadIdx.x >> 5;
  const int lane = threadIdx.x & 31;
  const int lr   = lane & 15;
  const int hi   = lane >> 4;
  const long row0 = (long)blockIdx.x * 16;
  const int  col  = wave * 16 + lr;

  // cooperative A-tile load: 16*K floats, all threads
  for (int i = threadIdx.x; i < 16 * K; i += blockDim.x) {
    int r = i / K, c = i % K;             // K power of two -> shifts
    As[r * PAD + c] = X[(row0 + r) * K + c];
  }
  __syncthreads();

  v8f c = {0.f, 0.f, 0.f, 0.f, 0.f, 0.f, 0.f, 0.f};
  const float* __restrict__ arow = &As[lr * PAD + 2 * hi];
  const float* __restrict__ wcol = W + (long)2 * hi * F + col;

#pragma unroll
  for (int k = 0; k < K; k += 4) {
    // A 16x4 layout: lanes 0-15 hold K=k,k+1 ; lanes 16-31 hold K=k+2,k+3
    v2f a = *(const v2f*)(arow + k);
    // B 4x16 layout: v0/v1 = rows k,k+1 (lanes 0-15) / k+2,k+3 (lanes 16-31)
    v2f b;
    b.x = wcol[(long)k * F];
    b.y = wcol[(long)(k + 1) * F];
    c = __builtin_amdgcn_wmma_f32_16x16x4_f32(false, a, false, b, (short)0, c, false, false);
  }

  // C/D layout: VGPR r -> row row0+r (lanes 0-15) / row0+8+r (lanes 16-31)
  float* __restrict__ out = H + (row0 + hi * 8) * F + col;
#pragma unroll
  for (int r = 0; r < 8; ++r) out[(long)r * F] = c[r];
}

// generic scalar GEMM fallback (also covers N % 16 tail rows)
__global__ void gcn_gemm_ref(const float* __restrict__ X, const float* __restrict__ W,
                             float* __restrict__ H, int K, int F, int rowStart, int N) {
  int idx = blockIdx.x * blockDim.x + threadIdx.x;
  int row = rowStart + idx / F;
  int col = idx % F;
  if (row < N) {
    float s = 0.f;
    for (int k = 0; k < K; ++k) s += X[(long)row * K + k] * W[(long)k * F + col];
    H[(long)row * F + col] = s;
  }
}

// ---------------- self-loop init: agg = h * dinv^2 (float4) ----------------
__global__ void gcn_self_loop4(const float4* __restrict__ h, const float* __restrict__ dinv,
                               float4* __restrict__ agg, int qshift, long totq) {
  long i = (long)blockIdx.x * blockDim.x + threadIdx.x;
  if (i < totq) {
    float d = dinv[i >> qshift];
    float s = d * d;
    float4 v = h[i];
    v.x *= s; v.y *= s; v.z *= s; v.w *= s;
    agg[i] = v;
  }
}

// ------- edge scatter: agg[dst] += h[src]*norm ; one thread per (edge, quad) -------
__global__ void gcn_scatter4(const float4* __restrict__ h, const float* __restrict__ dinv,
                             const int* __restrict__ src, const int* __restrict__ dst,
                             float* __restrict__ agg, int qshift, long totq) {
  long gid = (long)blockIdx.x * blockDim.x + threadIdx.x;
  if (gid < totq) {
    long e = gid >> qshift;
    int  q = (int)(gid & ((1 << qshift) - 1));
    int s = src[e], d = dst[e];
    float norm = dinv[s] * dinv[d];
    float4 v = h[((long)s << qshift) + q];      // coalesced global_load_b128
    float* ad = agg + ((((long)d << qshift) + q) << 2);
    atomic_add_f32_dev(ad + 0, v.x * norm);
    atomic_add_f32_dev(ad + 1, v.y * norm);
    atomic_add_f32_dev(ad + 2, v.z * norm);
    atomic_add_f32_dev(ad + 3, v.w * norm);
  }
}

// ---------------- bias + relu (float4) ----------------
__global__ void gcn_bias_relu4(const float4* __restrict__ agg, const float4* __restrict__ bias,
                               float4* __restrict__ out, int qshift, long totq) {
  long i = (long)blockIdx.x * blockDim.x + threadIdx.x;
  if (i < totq) {
    float4 v = agg[i];
    float4 b = bias[i & ((1 << qshift) - 1)];
    v.x += b.x; v.y += b.y; v.z += b.z; v.w += b.w;
    v.x = v.x > 0.f ? v.x : 0.f;
    v.y = v.y > 0.f ? v.y : 0.f;
    v.z = v.z > 0.f ? v.z : 0.f;
    v.w = v.w > 0.f ? v.w : 0.f;
    out[i] = v;
  }
}

static inline int ilog2(int x) { int s = 0; while ((1 << s) < x) ++s; return s; }

extern "C" void kernel_launch(void* const* d_in, const int* in_sizes, int n_in,
                              void* d_out, int out_size, void* d_ws, size_t ws_size,
                              hipStream_t stream) {
  const float* x  = (const float*)d_in[0];
  const int*   ei = (const int*)d_in[1];
  const float* W1 = (const float*)d_in[2];
  const float* b1 = (const float*)d_in[3];
  const float* W2 = (const float*)d_in[4];
  const float* b2 = (const float*)d_in[5];

  const int HID = in_sizes[3];          // 128
  const int IN  = in_sizes[2] / HID;    // 128
  const int OUT = in_sizes[5];          // 64
  const int N   = in_sizes[0] / IN;     // 100000
  const int E   = in_sizes[1] / 2;      // 1600000
  const int* src = ei;
  const int* dst = ei + E;

  // workspace: dinv [N] | bufh [N*HID] | bufa [N*HID]
  char* ws = (char*)d_ws;
  float* dinv = (float*)ws;
  size_t off    = ((size_t)N * sizeof(float) + 255) & ~(size_t)255;
  size_t hbytes = ((size_t)N * HID * sizeof(float) + 255) & ~(size_t)255;
  float* bufh = (float*)(ws + off);
  float* bufa = (float*)(ws + off + hbytes);

  const int T = 256;

  // degrees with self-loops -> dinv
  gcn_deg_init <<<(N + T - 1) / T, T, 0, stream>>>(dinv, N);
  gcn_deg_count<<<(E + T - 1) / T, T, 0, stream>>>(dst, dinv, E);
  gcn_deg_rsqrt<<<(N + T - 1) / T, T, 0, stream>>>(dinv, N);

  const int rt  = N / 16;
  const int rem = N % 16;
  const int q1 = ilog2(HID / 4);        // 5
  const int q2 = ilog2(OUT / 4);        // 4

  // ---- layer 1: h1 = x@W1 -> bufh ; agg1 -> bufa ; x1 = relu(agg1+b1) -> bufh ----
  if (IN == 128 && HID == 128 && rt > 0) {
    gcn_gemm_wmma<128, 128><<<rt, 256, 0, stream>>>(x, W1, bufh);
    if (rem)
      gcn_gemm_ref<<<(rem * HID + T - 1) / T, T, 0, stream>>>(x, W1, bufh, IN, HID, rt * 16, N);
  } else {
    gcn_gemm_ref<<<((long)N * HID + T - 1) / T, T, 0, stream>>>(x, W1, bufh, IN, HID, 0, N);
  }

  long totq1 = (long)N * (HID / 4);
  long edgq1 = (long)E * (HID / 4);
  gcn_self_loop4<<<(unsigned)((totq1 + T - 1) / T), T, 0, stream>>>(
      (const float4*)bufh, dinv, (float4*)bufa, q1, totq1);
  gcn_scatter4<<<(unsigned)((edgq1 + T - 1) / T), T, 0, stream>>>(
      (const float4*)bufh, dinv, src, dst, bufa, q1, edgq1);
  gcn_bias_relu4<<<(unsigned)((totq1 + T - 1) / T), T, 0, stream>>>(
      (const float4*)bufa, (const float4*)b1, (float4*)bufh, q1, totq1);

  // ---- layer 2: h2 = x1@W2 -> bufa ; agg2 -> bufh ; out = relu(agg2+b2) -> d_out ----
  if (HID == 128 && OUT == 64 && rt > 0) {
    gcn_gemm_wmma<128, 64><<<rt, 128, 0, stream>>>(bufh, W2, bufa);
    if (rem)
      gcn_gemm_ref<<<(rem * OUT + T - 1) / T, T, 0, stream>>>(bufh, W2, bufa, HID, OUT, rt * 16, N);
  } else {
    gcn_gemm_ref<<<((long)N * OUT + T - 1) / T, T, 0, stream>>>(bufh, W2, bufa, HID, OUT, 0, N);
  }

  long totq2 = (long)N * (OUT / 4);
  long edgq2 = (long)E * (OUT / 4);
  gcn_self_loop4<<<(unsigned)((totq2 + T - 1) / T), T, 0, stream>>>(
      (const float4*)bufa, dinv, (float4*)bufh, q2, totq2);
  gcn_scatter4<<<(unsigned)((edgq2 + T - 1) / T), T, 0, stream>>>(
      (const float4*)bufa, dinv, src, dst, bufh, q2, edgq2);
  gcn_bias_relu4<<<(unsigned)((totq2 + T - 1) / T), T, 0, stream>>>(
      (const float4*)bufh, (const float4*)b2, (float4*)d_out, q2, totq2);
}